// Temporal_dot_80951543595323
// MI455X (gfx1250) — compile-verified
//
#include <hip/hip_runtime.h>
#include <hip/hip_bf16.h>

// ---------------------------------------------------------------------------
// Temporal_dot for MI455X (gfx1250, wave32)
//   B=16 T=128 C=32 H=W=16  -> D=8192, F=16, G=4096, M=B*T=2048
//   Heavy op: v = x1 @ w3^T  (2048 x 4096 x 8192 fp32 GEMM, 137 GFLOP)
//   V_WMMA_F32_16X16X4_F32 + double-buffered GLOBAL_LOAD_ASYNC_TO_LDS_B128.
// ---------------------------------------------------------------------------

constexpr int nB = 16;      // batches
constexpr int nT = 128;     // time steps
constexpr int nD = 8192;    // C*H*W
constexpr int nF = 16;      // NUM_FEAT
constexpr int nG = 4096;    // NUM_FEAT*H*W
constexpr int nM = 2048;    // B*T

typedef float v2f __attribute__((ext_vector_type(2)));
typedef float v8f __attribute__((ext_vector_type(8)));

// fp32 WMMA: D(16x16,f32) = A(16x4,f32) * B(4x16,f32) + C
__device__ __forceinline__ v8f wmma_f32(v2f a, v2f b, v8f c) {
    return __builtin_amdgcn_wmma_f32_16x16x4_f32(
        false, a, false, b, (short)0, c, false, false);
}

// ---------------------------------------------------------------------------
// Async global -> LDS copy (16 bytes / lane), gfx1250 ASYNCcnt path.
// Builtin prototype (from hipcc diagnostic): arg0/arg1 are pointers to
// 16-byte int vectors.  Falls back to sync float4 staging if unavailable.
// ---------------------------------------------------------------------------
#if __has_builtin(__builtin_amdgcn_global_load_async_to_lds_b128)
#define TD_HAS_ASYNC 1
#else
#define TD_HAS_ASYNC 0
#endif

typedef int v4i_t __attribute__((vector_size(4 * sizeof(int))));
typedef __attribute__((address_space(1))) v4i_t* gv4i_p;   // global ptr to v4i
typedef __attribute__((address_space(3))) v4i_t* lv4i_p;   // LDS ptr to v4i

__device__ __forceinline__ void async_copy_b128(const float* gsrc, float* ldst) {
#if TD_HAS_ASYNC
    // generic->AS1 / generic->AS3 through integer round-trip:
    // low 32 bits of a generic LDS pointer are the LDS byte offset (ISA 10.2).
    __builtin_amdgcn_global_load_async_to_lds_b128(
        (gv4i_p)(unsigned long long)(uintptr_t)gsrc,
        (lv4i_p)(unsigned int)(uintptr_t)ldst,
        0, 0);
#else
    *(float4*)ldst = *(const float4*)gsrc;
#endif
}

__device__ __forceinline__ void wait_async_zero() {
#if TD_HAS_ASYNC
#if __has_builtin(__builtin_amdgcn_s_wait_asynccnt)
    __builtin_amdgcn_s_wait_asynccnt(0);
#else
    asm volatile("s_wait_asynccnt 0x0" ::: "memory");
#endif
#endif
}

// ---------------------------------------------------------------------------
// Kernel 1: q = x1*w1^T + b1 ; k = x2*w2^T + b2     (one block per (b,t) row)
// ---------------------------------------------------------------------------
__global__ __launch_bounds__(256)
void td_qk_kernel(const float* __restrict__ x1, const float* __restrict__ x2,
                  const float* __restrict__ w1, const float* __restrict__ b1,
                  const float* __restrict__ w2, const float* __restrict__ b2,
                  float* __restrict__ q, float* __restrict__ k) {
    const int row  = blockIdx.x;          // 0..M-1
    const int tid  = threadIdx.x;         // 0..255
    const int wave = tid >> 5;            // wave32
    const int lane = tid & 31;

    float accq[nF], acck[nF];
#pragma unroll
    for (int f = 0; f < nF; ++f) { accq[f] = 0.f; acck[f] = 0.f; }

    const float* xr1 = x1 + (size_t)row * nD;
    const float* xr2 = x2 + (size_t)row * nD;
    for (int d = tid; d < nD; d += 256) {
        const float a = xr1[d];
        const float c = xr2[d];
#pragma unroll
        for (int f = 0; f < nF; ++f) {
            accq[f] = fmaf(a, w1[f * nD + d], accq[f]);
            acck[f] = fmaf(c, w2[f * nD + d], acck[f]);
        }
    }
    // wave32 butterfly reduction
#pragma unroll
    for (int f = 0; f < nF; ++f) {
#pragma unroll
        for (int m = 16; m >= 1; m >>= 1) {
            accq[f] += __shfl_xor(accq[f], m, 32);
            acck[f] += __shfl_xor(acck[f], m, 32);
        }
    }
    __shared__ float lq[8][nF];
    __shared__ float lk[8][nF];
    if (lane == 0) {
#pragma unroll
        for (int f = 0; f < nF; ++f) { lq[wave][f] = accq[f]; lk[wave][f] = acck[f]; }
    }
    __syncthreads();
    if (tid < nF) {
        float s = b1[tid];
#pragma unroll
        for (int w = 0; w < 8; ++w) s += lq[w][tid];
        q[(size_t)row * nF + tid] = s;
    } else if (tid < 2 * nF) {
        const int f = tid - nF;
        float s = b2[f];
#pragma unroll
        for (int w = 0; w < 8; ++w) s += lk[w][f];
        k[(size_t)row * nF + f] = s;
    }
}

// ---------------------------------------------------------------------------
// Kernel 2: scores[b,t,s] = q[b,t,:].k[b,s,:], softmax over t (axis=1!)
// One 128-thread block per (s, b); thread index = t.
// ---------------------------------------------------------------------------
__global__ __launch_bounds__(128)
void td_softmax_kernel(const float* __restrict__ q, const float* __restrict__ k,
                       float* __restrict__ attn) {
    const int s = blockIdx.x;             // 0..T-1
    const int b = blockIdx.y;             // 0..B-1
    const int t = threadIdx.x;            // 0..T-1

    const float* qr = q + ((size_t)b * nT + t) * nF;
    const float* kr = k + ((size_t)b * nT + s) * nF;
    float val = 0.f;
#pragma unroll
    for (int f = 0; f < nF; ++f) val = fmaf(qr[f], kr[f], val);

    __shared__ float red[nT];
    red[t] = val;
    __syncthreads();
    for (int off = 64; off > 0; off >>= 1) {
        if (t < off) red[t] = fmaxf(red[t], red[t + off]);
        __syncthreads();
    }
    const float mx = red[0];
    __syncthreads();
    const float e = expf(val - mx);
    red[t] = e;
    __syncthreads();
    for (int off = 64; off > 0; off >>= 1) {
        if (t < off) red[t] += red[t + off];
        __syncthreads();
    }
    attn[((size_t)b * nT + t) * nT + s] = e / red[0];
}

// ---------------------------------------------------------------------------
// Kernel 3: v[m,g] = sum_d x1[m,d]*w3[g,d] + b3[g]
// M=2048, N=4096, K=8192, fp32 WMMA 16x16x4.
// Workgroup: 8 waves, 128(M) x 64(N) tile; K in LDS chunks of 32,
// DOUBLE-buffered, staged with async global->LDS b128 (ASYNCcnt).
// LDS row stride 36 floats: 16B-aligned stores, conflict-free b64 reads.
// ---------------------------------------------------------------------------
constexpr int KC     = 32;
constexpr int LDSTR  = 36;
constexpr int AELEMS = 128 * LDSTR;   // 4608 floats
constexpr int BELEMS = 64 * LDSTR;    // 2304 floats

__global__ __launch_bounds__(256)
void td_gemm_v_kernel(const float* __restrict__ x1, const float* __restrict__ w3,
                      const float* __restrict__ b3, float* __restrict__ v) {
    __shared__ float Alds[2 * AELEMS];    // 2 x 18 KB
    __shared__ float Blds[2 * BELEMS];    // 2 x  9 KB

    const int tid   = threadIdx.x;
    const int wave  = tid >> 5;
    const int lane  = tid & 31;
    const int nbase = blockIdx.x * 64;
    const int mbase = blockIdx.y * 128;

    const int rrow = lane & 15;           // fragment row within 16
    const int koff = (lane >> 4) << 1;    // lanes 16-31 hold K+2,K+3 (ISA 7.12.2)

    v8f acc[4];
#pragma unroll
    for (int j = 0; j < 4; ++j) acc[j] = (v8f)(0.f);

    // ---- stage one K-chunk (A: 128x32, B: 64x32) into LDS buffer `buf` ----
    auto stage = [&](int k0, int buf) {
        float* Ash = &Alds[buf * AELEMS];
        float* Bsh = &Blds[buf * BELEMS];
#pragma unroll
        for (int i = 0; i < 4; ++i) {
            const int idx = tid + 256 * i;          // 0..1023
            const int row = idx >> 3;
            const int c4  = (idx & 7) << 2;
            async_copy_b128(&x1[(size_t)(mbase + row) * nD + k0 + c4],
                            &Ash[row * LDSTR + c4]);
        }
#pragma unroll
        for (int i = 0; i < 2; ++i) {
            const int idx = tid + 256 * i;          // 0..511
            const int row = idx >> 3;
            const int c4  = (idx & 7) << 2;
            async_copy_b128(&w3[(size_t)(nbase + row) * nD + k0 + c4],
                            &Bsh[row * LDSTR + c4]);
        }
    };

    // prologue: fill buffer 0
    stage(0, 0);
    wait_async_zero();
    __syncthreads();

    int cur = 0;
    for (int k0 = 0; k0 < nD; k0 += KC) {
        const int nxt = cur ^ 1;
        if (k0 + KC < nD) stage(k0 + KC, nxt);     // overlap with compute below

        const float* Ash = &Alds[cur * AELEMS];
        const float* Bsh = &Blds[cur * BELEMS];
#pragma unroll
        for (int kk = 0; kk < KC; kk += 4) {
            const v2f a = *(const v2f*)&Ash[(wave * 16 + rrow) * LDSTR + kk + koff];
#pragma unroll
            for (int j = 0; j < 4; ++j) {
                const v2f bf = *(const v2f*)&Bsh[(j * 16 + rrow) * LDSTR + kk + koff];
                acc[j] = wmma_f32(a, bf, acc[j]);
            }
        }
        wait_async_zero();       // next buffer fully landed (this wave)
        __syncthreads();         // ... and across all waves; cur free to reuse
        cur = nxt;
    }

    // --- store with bias; C/D layout: VGPR i -> M = i + 8*(lane>=16), N = lane&15
    const int rsel = lane >> 4;
#pragma unroll
    for (int j = 0; j < 4; ++j) {
        const int col = nbase + j * 16 + rrow;
        const float bias = b3[col];
#pragma unroll
        for (int i = 0; i < 8; ++i) {
            const int rowm = mbase + wave * 16 + i + 8 * rsel;
            v[(size_t)rowm * nG + col] = acc[j][i] + bias;
        }
    }
}

// ---------------------------------------------------------------------------
// Kernel 4: out[b,t,g] = sum_s attn[b,t,s] * v[b,s,g]
// Per batch: 128 x 4096 x 128 GEMM. One wave per 16x64 tile, WMMA f32.
// ---------------------------------------------------------------------------
__global__ __launch_bounds__(32)
void td_av_kernel(const float* __restrict__ attn, const float* __restrict__ v,
                  float* __restrict__ out) {
    const int n0 = blockIdx.x * 64;
    const int t0 = blockIdx.y * 16;
    const int b  = blockIdx.z;
    const int lane = threadIdx.x;
    const int rrow = lane & 15;
    const int koff = (lane >> 4) << 1;
    const int rsel = lane >> 4;

    v8f acc[4];
#pragma unroll
    for (int j = 0; j < 4; ++j) acc[j] = (v8f)(0.f);

    const float* A = attn + (size_t)b * nT * nT;   // [T][T] row-major
    const float* V = v + (size_t)b * nT * nG;      // [T][G] row-major

    for (int kb = 0; kb < nT; kb += 4) {
        const v2f a = *(const v2f*)&A[(size_t)(t0 + rrow) * nT + kb + koff];
#pragma unroll
        for (int j = 0; j < 4; ++j) {
            const int col = n0 + j * 16 + rrow;
            v2f bf;
            bf.x = V[(size_t)(kb + koff) * nG + col];       // B[k][n]   = v[b, k, n]
            bf.y = V[(size_t)(kb + koff + 1) * nG + col];   // B[k+1][n]
            acc[j] = wmma_f32(a, bf, acc[j]);
        }
    }
#pragma unroll
    for (int j = 0; j < 4; ++j) {
        const int col = n0 + j * 16 + rrow;
#pragma unroll
        for (int i = 0; i < 8; ++i) {
            const int rowt = t0 + i + 8 * rsel;
            out[((size_t)b * nT + rowt) * nG + col] = acc[j][i];
        }
    }
}

// ---------------------------------------------------------------------------
extern "C" void kernel_launch(void* const* d_in, const int* in_sizes, int n_in,
                              void* d_out, int out_size, void* d_ws, size_t ws_size,
                              hipStream_t stream) {
    const float* x1 = (const float*)d_in[0];
    const float* x2 = (const float*)d_in[1];
    const float* w1 = (const float*)d_in[2];
    const float* b1 = (const float*)d_in[3];
    const float* w2 = (const float*)d_in[4];
    const float* b2 = (const float*)d_in[5];
    const float* w3 = (const float*)d_in[6];
    const float* b3 = (const float*)d_in[7];
    float* out = (float*)d_out;

    // workspace partition (floats): q | k | attn | v   (~33.3 MB total)
    float* q    = (float*)d_ws;
    float* kbuf = q    + (size_t)nM * nF;         // 32768
    float* attn = kbuf + (size_t)nM * nF;         // 32768
    float* v    = attn + (size_t)nB * nT * nT;    // 262144

    td_qk_kernel<<<nM, 256, 0, stream>>>(x1, x2, w1, b1, w2, b2, q, kbuf);
    td_softmax_kernel<<<dim3(nT, nB), 128, 0, stream>>>(q, kbuf, attn);
    td_gemm_v_kernel<<<dim3(nG / 64, nM / 128), 256, 0, stream>>>(x1, w3, b3, v);
    td_av_kernel<<<dim3(nG / 64, nT / 16, nB), 32, 0, stream>>>(attn, v, out);
}